// CSDraftingEncoderDecoderModel_9655086482171
// MI455X (gfx1250) — compile-verified
//
#include <hip/hip_runtime.h>
#include <cstdint>
#include <math.h>

// ---------------------------------------------------------------------------
// Speculative-decode verification (CSDrafting): B=1, L=520, V=32128, r=512.
// Memory-bound: ~135 MB traffic, ~5.8us at 23.3 TB/s. No GEMM -> no WMMA;
// the relevant CDNA5 path is async global<->LDS streaming (ASYNCcnt).
// Outputs (flat, float32): id_res[L+1] | prob_res[L*V] | n_match[1]
// ---------------------------------------------------------------------------

#define TPB    256
#define CHUNK  4096   // floats per block in the streaming phase (16 KB)
#define MAXROWS 32    // stats slots (K+1 == 9 in this setup)

typedef int   v4i __attribute__((vector_size(16)));      // matches builtin sig
typedef float v4f __attribute__((ext_vector_type(4)));   // for NT vector loads

#if defined(__HIP_DEVICE_COMPILE__) && defined(__gfx1250__) &&                 \
    __has_builtin(__builtin_amdgcn_global_load_async_to_lds_b128) &&           \
    __has_builtin(__builtin_amdgcn_global_store_async_from_lds_b32) &&         \
    __has_builtin(__builtin_amdgcn_s_wait_asynccnt)
#define USE_ASYNC_LDS_COPY 1
#else
#define USE_ASYNC_LDS_COPY 0
#endif

#if USE_ASYNC_LDS_COPY
#define GLOBAL_AS __attribute__((address_space(1)))
#define LDS_AS    __attribute__((address_space(3)))
#endif

// ---------------------------------------------------------------------------
// Phase 1: per-row max / first-argmax / sum(exp(x-max)) for rows r-1 .. L-1.
// One block per position; blocks below r-1 exit immediately.
// ---------------------------------------------------------------------------
__global__ void __launch_bounds__(TPB)
row_stats_kernel(const float* __restrict__ logits,
                 const int* __restrict__ p_r,
                 int L, int V,
                 float* __restrict__ rowmax,
                 float* __restrict__ rowsum,
                 int* __restrict__ rowarg) {
  const int r   = *p_r;
  const int pos = blockIdx.x;               // 0 .. L-1
  if (pos < r - 1) return;
  const int row = pos - (r - 1);            // 0 .. K
  if (row >= MAXROWS) return;
  const float* __restrict__ x = logits + (size_t)pos * (size_t)V;
  const int tid = threadIdx.x;

  __shared__ float smax[TPB];
  __shared__ int   sidx[TPB];
  __shared__ float ssum[TPB];

  // pass 1: max + first-occurrence argmax
  float m  = -INFINITY;
  int   mi = V;
  for (int i = tid; i < V; i += TPB) {
    float v = x[i];
    if (v > m) { m = v; mi = i; }           // ascending i => first occurrence
  }
  smax[tid] = m; sidx[tid] = mi;
  __syncthreads();
  for (int s = TPB / 2; s > 0; s >>= 1) {
    if (tid < s) {
      float om = smax[tid + s]; int oi = sidx[tid + s];
      float cm = smax[tid];     int ci = sidx[tid];
      if (om > cm || (om == cm && oi < ci)) { smax[tid] = om; sidx[tid] = oi; }
    }
    __syncthreads();
  }
  const float rm = smax[0];
  const int   ra = sidx[0];

  // pass 2: sum of exp(x - max)
  float acc = 0.0f;
  for (int i = tid; i < V; i += TPB) acc += expf(x[i] - rm);
  ssum[tid] = acc;
  __syncthreads();
  for (int s = TPB / 2; s > 0; s >>= 1) {
    if (tid < s) ssum[tid] += ssum[tid + s];
    __syncthreads();
  }
  if (tid == 0) {
    rowmax[row] = rm;
    rowsum[row] = ssum[0];
    rowarg[row] = ra;
  }
}

// ---------------------------------------------------------------------------
// Phase 2: accept chain -> n_match; write id_res[0..L] and n_match scalar.
// ---------------------------------------------------------------------------
__global__ void __launch_bounds__(TPB)
decide_kernel(const float* __restrict__ logits,
              const float* __restrict__ probs,
              const int* __restrict__ ids,
              const int* __restrict__ p_r,
              const int* __restrict__ p_len,
              const float* __restrict__ rowmax,
              const float* __restrict__ rowsum,
              const int* __restrict__ rowarg,
              int L, int V,
              float* __restrict__ out,       // full output base
              int* __restrict__ p_nmatch) {
  const int r = *p_r;
  const int K = L - r;
  __shared__ int sh_nm;

  if (threadIdx.x == 0) {
    const int len = *p_len;
    int nm = 0;
    for (int j = 0; j < K; ++j) {
      const int   draft = ids[r + j];
      const int   tgt   = rowarg[j];              // argmax at position r-1+j
      const bool  eq    = (tgt == draft);
      const size_t off  = (size_t)(r - 1 + j) * (size_t)V + (size_t)draft;
      const float tp = expf(logits[off] - rowmax[j]) / rowsum[j];
      const float pp = probs[off];
      const bool lenient = (len > 1) && (tp > pp / (float)len);
      if (eq || lenient) nm++; else break;
    }
    sh_nm = nm;
    *p_nmatch = nm;
    out[(size_t)(L + 1) + (size_t)L * (size_t)V] = (float)nm;  // n_match output
  }
  __syncthreads();
  const int nm = sh_nm;

  // id_res[0..r-1] = input_ids[0..r-1]
  for (int i = threadIdx.x; i < r; i += TPB) out[i] = (float)ids[i];
  // id_res[r..r+K]: draft while accepted, bonus target token at nm, else 0
  for (int q = threadIdx.x; q <= K; q += TPB) {
    float v;
    if (q < nm)       v = (float)ids[r + q];
    else if (q == nm) v = (float)rowarg[q];
    else              v = 0.0f;
    out[r + q] = v;
  }
}

// ---------------------------------------------------------------------------
// Phase 3: stream prob_res. Rows < r: copy probs (async global->LDS->global
// DMA on CDNA5; NT vector fallback). Rows >= r: masked softmax or zeros.
// Output region starts at byte offset (L+1)*4 == 2084 (= 4 mod 16), so global
// stores are 4B-aligned B32; loads stay 16B-aligned B128.
// ---------------------------------------------------------------------------
__global__ void __launch_bounds__(TPB)
stream_probs_kernel(const float* __restrict__ probs,
                    const float* __restrict__ logits,
                    const int* __restrict__ p_r,
                    const float* __restrict__ rowmax,
                    const float* __restrict__ rowsum,
                    const int* __restrict__ p_nmatch,
                    int L, int V, int chunksPerRow,
                    float* __restrict__ outp) {   // = out + (L+1)
  const int bx   = blockIdx.x;
  const int row  = bx / chunksPerRow;
  const int base = (bx - row * chunksPerRow) * CHUNK;
  const int tid  = threadIdx.x;
  const int r    = *p_r;
  const int rem  = V - base;
  const int n    = rem < CHUNK ? rem : CHUNK;
  float* __restrict__ dst = outp + (size_t)row * (size_t)V + base;

  if (row < r) {
    const float* __restrict__ src = probs + (size_t)row * (size_t)V + base;
#if USE_ASYNC_LDS_COPY
    __shared__ __align__(16) float lbuf[CHUNK];
    if (n == CHUNK && ((((uintptr_t)src) & 15u) == 0)) {
      // Per-wave DMA: each thread owns one 16B slot per round; no cross-wave
      // dependency, so ASYNCcnt waits alone order load->store correctly.
#pragma unroll
      for (int rnd = 0; rnd < CHUNK / (TPB * 4); ++rnd) {
        const int off = rnd * (TPB * 4) + tid * 4;
        __builtin_amdgcn_global_load_async_to_lds_b128(
            (GLOBAL_AS v4i*)(src + off),
            (LDS_AS v4i*)(lbuf + off), 0, 0);
      }
      __builtin_amdgcn_s_wait_asynccnt(0);
#pragma unroll
      for (int rnd = 0; rnd < CHUNK / (TPB * 4); ++rnd) {
        const int off = rnd * (TPB * 4) + tid * 4;
#pragma unroll
        for (int k = 0; k < 4; ++k) {
          __builtin_amdgcn_global_store_async_from_lds_b32(
              (GLOBAL_AS int*)(dst + off + k),
              (LDS_AS int*)(lbuf + off + k), 0, 0);
        }
      }
      __builtin_amdgcn_s_wait_asynccnt(0);   // LDS is freed at block exit
      return;
    }
#endif
    // Fallback: B128 NT loads + B32 NT stores.
    if ((((uintptr_t)src) & 15u) == 0) {
      const int n4 = n >> 2;
      const v4f* __restrict__ s4 = (const v4f*)src;
      for (int i = tid; i < n4; i += TPB) {
        v4f v = __builtin_nontemporal_load(s4 + i);
        __builtin_nontemporal_store(v.x, dst + 4 * i + 0);
        __builtin_nontemporal_store(v.y, dst + 4 * i + 1);
        __builtin_nontemporal_store(v.z, dst + 4 * i + 2);
        __builtin_nontemporal_store(v.w, dst + 4 * i + 3);
      }
      for (int i = (n4 << 2) + tid; i < n; i += TPB)
        __builtin_nontemporal_store(src[i], dst + i);
    } else {
      for (int i = tid; i < n; i += TPB) dst[i] = src[i];
    }
    return;
  }

  // Speculative-tail rows: new_probs[j] = (j < n_match) ? softmax row : 0
  const int j  = row - r;
  const int nm = *p_nmatch;
  if (j < nm && j < MAXROWS) {
    const float rm = rowmax[j];
    const float rs = rowsum[j];
    const float* __restrict__ src =
        logits + (size_t)(r - 1 + j) * (size_t)V + base;
    for (int i = tid; i < n; i += TPB) {
      const float v = expf(src[i] - rm) / rs;
      __builtin_nontemporal_store(v, dst + i);
    }
  } else {
    for (int i = tid; i < n; i += TPB)
      __builtin_nontemporal_store(0.0f, dst + i);
  }
}

// ---------------------------------------------------------------------------
// Host launcher. Inputs: target_logits(f32 L*V), probs(f32 L*V),
// input_ids(i32 L), review_index(i32 1), leniency(i32 1).
// ---------------------------------------------------------------------------
extern "C" void kernel_launch(void* const* d_in, const int* in_sizes, int n_in,
                              void* d_out, int out_size, void* d_ws, size_t ws_size,
                              hipStream_t stream) {
  const float* target_logits = (const float*)d_in[0];
  const float* probs         = (const float*)d_in[1];
  const int*   input_ids     = (const int*)d_in[2];
  const int*   p_r           = (const int*)d_in[3];
  const int*   p_len         = (const int*)d_in[4];

  const int L = in_sizes[2];                 // B == 1
  const int V = in_sizes[0] / (L > 0 ? L : 1);

  float* out = (float*)d_out;

  // Workspace layout (floats): [0..31] rowmax, [32..63] rowsum,
  // [64..95] rowarg (as int), [96] n_match (as int).
  float* wsf     = (float*)d_ws;
  float* rowmax  = wsf;
  float* rowsum  = wsf + MAXROWS;
  int*   rowarg  = (int*)(wsf + 2 * MAXROWS);
  int*   nmatch  = (int*)(wsf + 3 * MAXROWS);

  // Phase 1: row reductions (only K+1 blocks do work; rest exit)
  row_stats_kernel<<<L, TPB, 0, stream>>>(target_logits, p_r, L, V,
                                          rowmax, rowsum, rowarg);

  // Phase 2: accept chain + id_res + n_match
  decide_kernel<<<1, TPB, 0, stream>>>(target_logits, probs, input_ids,
                                       p_r, p_len, rowmax, rowsum, rowarg,
                                       L, V, out, nmatch);

  // Phase 3: stream prob_res (bulk copy + masked softmax tail)
  const int chunksPerRow = (V + CHUNK - 1) / CHUNK;
  stream_probs_kernel<<<L * chunksPerRow, TPB, 0, stream>>>(
      probs, target_logits, p_r, rowmax, rowsum, nmatch,
      L, V, chunksPerRow, out + (L + 1));
}